// V_network_71846212927506
// MI455X (gfx1250) — compile-verified
//
#include <hip/hip_runtime.h>
#include <hip/hip_bf16.h>

typedef __attribute__((ext_vector_type(2))) float v2f;
typedef __attribute__((ext_vector_type(8))) float v8f;

#define DMODEL 256
#define NAG 16
#define BT_TOTAL 800
#define NROWS (BT_TOTAL * NAG)   // 12800
static_assert(NROWS % 16 == 0, "");

__device__ __forceinline__ float geluf(float x) {
    // exact gelu: x * 0.5 * (1 + erf(x / sqrt(2)))
    return 0.5f * x * (1.0f + erff(x * 0.70710678118654752440f));
}

__device__ __forceinline__ v8f wmma4(v2f a, v2f b, v8f c) {
    // V_WMMA_F32_16X16X4_F32 : D = A(16x4 f32) x B(4x16 f32) + C(16x16 f32)
    return __builtin_amdgcn_wmma_f32_16x16x4_f32(
        /*neg_a=*/false, a, /*neg_b=*/false, b,
        /*c_mod=*/(short)0, c, /*reuse_a=*/false, /*reuse_b=*/false);
}

// ---------------------------------------------------------------------------
// Kernel 1: state/enemy embedding + gelu + enemy sum + agent emb + LN0.
// One block (256 threads) per bt sample. Outputs SE, SAE = SE + action_emb,
// AE = action_emb[acts].
// ---------------------------------------------------------------------------
__global__ __launch_bounds__(256) void embed_kernel(
    const float* __restrict__ states,        // [BT, 16, 60]
    const float* __restrict__ enemy_states,  // [BT, 16, 40]
    const int*   __restrict__ actions,       // [BT, 16]
    const float* __restrict__ agent_emb,     // [16, 256]
    const float* __restrict__ action_emb,    // [22, 256]
    const float* __restrict__ enemy_emb,     // [16, 256]
    const float* __restrict__ W_ally, const float* __restrict__ b_ally,   // [60,256],[256]
    const float* __restrict__ W_enemy, const float* __restrict__ b_enemy, // [40,256],[256]
    const float* __restrict__ ln0_g, const float* __restrict__ ln0_b,
    float* __restrict__ SE, float* __restrict__ SAE, float* __restrict__ AE)
{
    const int bt = blockIdx.x;
    const int d  = threadIdx.x;            // 0..255 -> output channel
    __shared__ float xbuf[NAG * DMODEL];
    __shared__ float mu[NAG], rs[NAG];

    // enemy contribution: sum_e gelu(enemy_states[e] @ W_enemy + b)[d] + enemy_emb[e][d]
    float esum = 0.0f;
    const float* es = enemy_states + (size_t)bt * 16 * 40;
    for (int e = 0; e < 16; ++e) {
        float t = b_enemy[d];
        for (int c = 0; c < 40; ++c) t += es[e * 40 + c] * W_enemy[c * DMODEL + d];
        esum += geluf(t) + enemy_emb[e * DMODEL + d];
    }

    const float* st = states + (size_t)bt * NAG * 60;
    for (int a = 0; a < NAG; ++a) {
        float t = b_ally[d];
        for (int c = 0; c < 60; ++c) t += st[a * 60 + c] * W_ally[c * DMODEL + d];
        xbuf[a * DMODEL + d] = geluf(t) + agent_emb[a * DMODEL + d] + esum;
    }
    __syncthreads();

    if (d < NAG) {
        float s = 0.0f, s2 = 0.0f;
        for (int c = 0; c < DMODEL; ++c) {
            float v = xbuf[d * DMODEL + c];
            s += v; s2 += v * v;
        }
        float m = s * (1.0f / DMODEL);
        mu[d] = m;
        rs[d] = rsqrtf(s2 * (1.0f / DMODEL) - m * m + 1e-5f);
    }
    __syncthreads();

    for (int a = 0; a < NAG; ++a) {
        float v = (xbuf[a * DMODEL + d] - mu[a]) * rs[a] * ln0_g[d] + ln0_b[d];
        size_t ro = ((size_t)bt * NAG + a) * DMODEL + d;
        int act = actions[bt * NAG + a];
        float ae = action_emb[(size_t)act * DMODEL + d];
        SE[ro]  = v;
        SAE[ro] = v + ae;
        AE[ro]  = ae;
    }
}

// ---------------------------------------------------------------------------
// Generic WMMA row-tile GEMM:  Y[M,256] = act(X[M,256] @ W[256,256] + bias)
// One block = 512 threads = 16 waves; block handles one 16-row M tile,
// wave w handles N tile w. K loop = 64 x V_WMMA_F32_16X16X4_F32.
// ---------------------------------------------------------------------------
__global__ __launch_bounds__(512) void gemm_wmma(
    const float* __restrict__ X, const float* __restrict__ W,
    const float* __restrict__ bias, float* __restrict__ Y, int doGelu)
{
    __shared__ float xs[16 * DMODEL];
    const int mtile = blockIdx.x;
    const float* Xrow = X + (size_t)mtile * 16 * DMODEL;
    for (int idx = threadIdx.x; idx < 16 * DMODEL / 4; idx += 512)
        ((float4*)xs)[idx] = ((const float4*)Xrow)[idx];
    __syncthreads();

    const int wave  = threadIdx.x >> 5;
    const int lane  = threadIdx.x & 31;
    const int ntile = wave;
    const int row   = lane & 15;
    const int khalf = lane >> 4;
    const int col   = ntile * 16 + (lane & 15);

    v8f acc = {};
    for (int kk = 0; kk < DMODEL; kk += 4) {
        const int k0 = kk + khalf * 2;
        v2f a; a.x = xs[row * DMODEL + k0];        a.y = xs[row * DMODEL + k0 + 1];
        v2f b; b.x = W[(size_t)k0 * DMODEL + col]; b.y = W[(size_t)(k0 + 1) * DMODEL + col];
        acc = wmma4(a, b, acc);
    }
    const float bc = bias ? bias[col] : 0.0f;
#pragma unroll
    for (int r = 0; r < 8; ++r) {
        const int ro = r + 8 * khalf;
        float v = acc[r] + bc;
        if (doGelu) v = geluf(v);
        Y[((size_t)mtile * 16 + ro) * DMODEL + col] = v;
    }
}

// ---------------------------------------------------------------------------
// Kernel 3: per-sample attention. One block (256 threads = 8 waves) per bt.
//   score = q @ k^T / 16            (WMMA, wave 0)
//   qh/kh = q,k projections via W_h (waves 1,2)
//   Gumbel hard gate, MASK arithmetic, range temperature, softmax, fw
//   agg   = fw @ Vfull - diag(fw)*AEV  (WMMA, 8 waves)
//   x     = LN1(oae_sum + agg)
// ---------------------------------------------------------------------------
__global__ __launch_bounds__(256) void attn_kernel(
    const float* __restrict__ Q, const float* __restrict__ K,
    const float* __restrict__ Vf, const float* __restrict__ AEV,
    const float* __restrict__ SE, const float* __restrict__ SAE,
    const float* __restrict__ agent_masks,   // [BT,16]
    const float* __restrict__ gumbel,        // [BT,1,16,16,2]
    const float* __restrict__ W_h, const float* __restrict__ b_h,   // [512,2],[2]
    const float* __restrict__ ln1_g, const float* __restrict__ ln1_b,
    float* __restrict__ X1)
{
    const int bt = blockIdx.x;
    const size_t base = (size_t)bt * NAG * DMODEL;
    __shared__ float qs[NAG * DMODEL], ks[NAG * DMODEL], xb[NAG * DMODEL];
    __shared__ float scoreb[NAG][NAG], attnb[NAG][NAG], svb[NAG][NAG];
    __shared__ float eb[NAG][NAG], fwb[NAG][NAG];
    __shared__ float qh[NAG][2], kh[NAG][2];
    __shared__ float denomr[NAG], em[NAG], ssum[NAG];
    __shared__ float saesum[DMODEL], mu[NAG], rs[NAG];

    for (int idx = threadIdx.x; idx < NAG * DMODEL / 4; idx += 256) {
        ((float4*)qs)[idx] = ((const float4*)(Q + base))[idx];
        ((float4*)ks)[idx] = ((const float4*)(K + base))[idx];
    }
    {   // column sums of SAE (for oae_sum)
        const int d = threadIdx.x;
        float s = 0.0f;
        for (int a = 0; a < NAG; ++a) s += SAE[base + a * DMODEL + d];
        saesum[d] = s;
    }
    __syncthreads();

    const int wave = threadIdx.x >> 5, lane = threadIdx.x & 31;
    if (wave == 0) {
        // score = q @ k^T / sqrt(256)
        const int row = lane & 15, khalf = lane >> 4, col = lane & 15;
        v8f acc = {};
        for (int kk = 0; kk < DMODEL; kk += 4) {
            const int k0 = kk + khalf * 2;
            v2f a; a.x = qs[row * DMODEL + k0]; a.y = qs[row * DMODEL + k0 + 1];
            v2f b; b.x = ks[col * DMODEL + k0]; b.y = ks[col * DMODEL + k0 + 1];
            acc = wmma4(a, b, acc);
        }
#pragma unroll
        for (int r = 0; r < 8; ++r) scoreb[r + 8 * khalf][col] = acc[r] * (1.0f / 16.0f);
    } else if (wave == 1) {
        const int i = lane >> 1, c = lane & 1;
        float t = 0.0f;
        for (int d0 = 0; d0 < DMODEL; ++d0) t += qs[i * DMODEL + d0] * W_h[d0 * 2 + c];
        qh[i][c] = t;
    } else if (wave == 2) {
        const int i = lane >> 1, c = lane & 1;
        float t = 0.0f;
        for (int d0 = 0; d0 < DMODEL; ++d0) t += ks[i * DMODEL + d0] * W_h[(DMODEL + d0) * 2 + c];
        kh[i][c] = t;
    }
    __syncthreads();

    const float MASKV = -3.40282346638528859812e+38f;  // finfo(f32).min
    const int i = threadIdx.x >> 4, j = threadIdx.x & 15;
    {
        const size_t gbase = ((size_t)bt * 256 + i * 16 + j) * 2;
        float l0 = qh[i][0] + kh[j][0] + b_h[0] + gumbel[gbase + 0];
        float l1 = qh[i][1] + kh[j][1] + b_h[1] + gumbel[gbase + 1];
        float hard = (l1 > l0) ? 1.0f : 0.0f;           // argmax, first-on-tie
        if (i == j) hard = 1.0f;                        // self edge forced on
        float ami = agent_masks[bt * NAG + i], amj = agent_masks[bt * NAG + j];
        bool dead = (amj == 0.0f) || (ami == 0.0f) || (i == j);
        float attnv = (dead ? MASKV : (1.0f - amj)) + (1.0f - hard) * MASKV; // may -> -inf
        float valid = (attnv != MASKV) ? 1.0f : 0.0f;
        attnb[i][j] = attnv;
        svb[i][j]   = scoreb[i][j] * valid;
    }
    __syncthreads();
    if (threadIdx.x < NAG) {
        float mx = svb[threadIdx.x][0], mn = mx;
        for (int jj = 1; jj < NAG; ++jj) {
            float v = svb[threadIdx.x][jj];
            mx = fmaxf(mx, v); mn = fminf(mn, v);
        }
        denomr[threadIdx.x] = mx - mn + 1e-5f;
    }
    __syncthreads();
    eb[i][j] = scoreb[i][j] / denomr[i] + attnb[i][j];
    __syncthreads();
    if (threadIdx.x < NAG) {
        float mx = eb[threadIdx.x][0];
        for (int jj = 1; jj < NAG; ++jj) mx = fmaxf(mx, eb[threadIdx.x][jj]);
        float s = 0.0f;
        for (int jj = 0; jj < NAG; ++jj) s += expf(eb[threadIdx.x][jj] - mx);
        em[threadIdx.x] = mx; ssum[threadIdx.x] = s;
    }
    __syncthreads();
    {
        float w  = expf(eb[i][j] - em[i]) / ssum[i];
        float fw = (i == j) ? 1.0f : w;
        float ami = agent_masks[bt * NAG + i], amj = agent_masks[bt * NAG + j];
        fwb[i][j] = fw * ami * amj;
    }
    __syncthreads();

    // agg = fw @ Vfull (K=16) ; epilogue applies diag correction + oae_sum
    {
        const int row = lane & 15, khalf = lane >> 4;
        for (int nt = wave * 2; nt < wave * 2 + 2; ++nt) {
            const int col = nt * 16 + (lane & 15);
            v8f acc = {};
            for (int kk = 0; kk < NAG; kk += 4) {
                const int k0 = kk + khalf * 2;
                v2f a; a.x = fwb[row][k0];                      a.y = fwb[row][k0 + 1];
                v2f b; b.x = Vf[base + (size_t)k0 * DMODEL + col];
                       b.y = Vf[base + (size_t)(k0 + 1) * DMODEL + col];
                acc = wmma4(a, b, acc);
            }
#pragma unroll
            for (int r = 0; r < 8; ++r) {
                const int ro = r + 8 * khalf;
                float agg = acc[r] - fwb[ro][ro] * AEV[base + (size_t)ro * DMODEL + col];
                float ae  = SAE[base + (size_t)ro * DMODEL + col]
                          - SE [base + (size_t)ro * DMODEL + col];
                xb[ro * DMODEL + col] = saesum[col] - ae + agg;   // oae_sum + agg
            }
        }
    }
    __syncthreads();
    if (threadIdx.x < NAG) {
        float s = 0.0f, s2 = 0.0f;
        for (int c = 0; c < DMODEL; ++c) {
            float v = xb[threadIdx.x * DMODEL + c];
            s += v; s2 += v * v;
        }
        float m = s * (1.0f / DMODEL);
        mu[threadIdx.x] = m;
        rs[threadIdx.x] = rsqrtf(s2 * (1.0f / DMODEL) - m * m + 1e-5f);
    }
    __syncthreads();
    {
        const int d = threadIdx.x;
        for (int a = 0; a < NAG; ++a)
            X1[base + a * DMODEL + d] =
                (xb[a * DMODEL + d] - mu[a]) * rs[a] * ln1_g[d] + ln1_b[d];
    }
}

// ---------------------------------------------------------------------------
// Residual + LayerNorm:  Y[row] = LN(A[row] + Bres[row]; g, b). Block per row.
// ---------------------------------------------------------------------------
__global__ __launch_bounds__(256) void resln_kernel(
    const float* __restrict__ A, const float* __restrict__ Bres,
    const float* __restrict__ g, const float* __restrict__ b,
    float* __restrict__ Y)
{
    const int row = blockIdx.x, d = threadIdx.x;
    __shared__ float sdata[DMODEL];
    float v = A[(size_t)row * DMODEL + d] + Bres[(size_t)row * DMODEL + d];
    sdata[d] = v; __syncthreads();
    for (int s = 128; s > 0; s >>= 1) { if (d < s) sdata[d] += sdata[d + s]; __syncthreads(); }
    float mean = sdata[0] * (1.0f / DMODEL); __syncthreads();
    float diff = v - mean;
    sdata[d] = diff * diff; __syncthreads();
    for (int s = 128; s > 0; s >>= 1) { if (d < s) sdata[d] += sdata[d + s]; __syncthreads(); }
    float rstd = rsqrtf(sdata[0] * (1.0f / DMODEL) + 1e-5f);
    Y[(size_t)row * DMODEL + d] = diff * rstd * g[d] + b[d];
}

// ---------------------------------------------------------------------------
// Final: v[row] = LN(XC[row]; ln3) @ W_out + b_out. Block per row.
// ---------------------------------------------------------------------------
__global__ __launch_bounds__(256) void head_kernel(
    const float* __restrict__ XC,
    const float* __restrict__ g, const float* __restrict__ b,
    const float* __restrict__ W_out, const float* __restrict__ b_out,
    float* __restrict__ out)
{
    const int row = blockIdx.x, d = threadIdx.x;
    __shared__ float sdata[DMODEL];
    float v = XC[(size_t)row * DMODEL + d];
    sdata[d] = v; __syncthreads();
    for (int s = 128; s > 0; s >>= 1) { if (d < s) sdata[d] += sdata[d + s]; __syncthreads(); }
    float mean = sdata[0] * (1.0f / DMODEL); __syncthreads();
    float diff = v - mean;
    sdata[d] = diff * diff; __syncthreads();
    for (int s = 128; s > 0; s >>= 1) { if (d < s) sdata[d] += sdata[d + s]; __syncthreads(); }
    float rstd = rsqrtf(sdata[0] * (1.0f / DMODEL) + 1e-5f);
    __syncthreads();
    sdata[d] = (diff * rstd * g[d] + b[d]) * W_out[d]; __syncthreads();
    for (int s = 128; s > 0; s >>= 1) { if (d < s) sdata[d] += sdata[d + s]; __syncthreads(); }
    if (d == 0) out[row] = sdata[0] + b_out[0];
}

// ---------------------------------------------------------------------------
extern "C" void kernel_launch(void* const* d_in, const int* in_sizes, int n_in,
                              void* d_out, int out_size, void* d_ws, size_t ws_size,
                              hipStream_t stream)
{
    const float* states       = (const float*)d_in[0];
    const float* enemy_states = (const float*)d_in[1];
    const int*   actions      = (const int*)  d_in[2];
    // d_in[3] rnn_hidden_state unused
    const float* agent_masks  = (const float*)d_in[4];
    const float* gumbel       = (const float*)d_in[5];
    const float* agent_emb    = (const float*)d_in[6];
    const float* action_emb   = (const float*)d_in[7];
    const float* enemy_emb    = (const float*)d_in[8];
    const float* W_ally  = (const float*)d_in[9];  const float* b_ally  = (const float*)d_in[10];
    const float* W_enemy = (const float*)d_in[11]; const float* b_enemy = (const float*)d_in[12];
    const float* ln0_g   = (const float*)d_in[13]; const float* ln0_b   = (const float*)d_in[14];
    const float* W_k     = (const float*)d_in[15]; const float* b_k     = (const float*)d_in[16];
    const float* W_q     = (const float*)d_in[17]; const float* b_q     = (const float*)d_in[18];
    const float* W_v     = (const float*)d_in[19]; const float* b_v     = (const float*)d_in[20];
    const float* W_h     = (const float*)d_in[21]; const float* b_h     = (const float*)d_in[22];
    const float* ln1_g   = (const float*)d_in[23]; const float* ln1_b   = (const float*)d_in[24];
    const float* W_f1    = (const float*)d_in[25]; const float* b_f1    = (const float*)d_in[26];
    const float* W_f2    = (const float*)d_in[27]; const float* b_f2    = (const float*)d_in[28];
    const float* ln2_g   = (const float*)d_in[29]; const float* ln2_b   = (const float*)d_in[30];
    const float* W_c     = (const float*)d_in[31]; const float* b_c     = (const float*)d_in[32];
    const float* ln3_g   = (const float*)d_in[33]; const float* ln3_b   = (const float*)d_in[34];
    const float* W_out   = (const float*)d_in[35]; const float* b_out   = (const float*)d_in[36];

    float* ws = (float*)d_ws;
    const size_t NB = (size_t)NROWS * DMODEL;
    float* SE  = ws + 0 * NB;
    float* SAE = ws + 1 * NB;
    float* AE  = ws + 2 * NB;   // reused as X1 after AEV is computed
    float* Qb  = ws + 3 * NB;   // reused as F1
    float* Kb  = ws + 4 * NB;   // reused as F2
    float* Vfb = ws + 5 * NB;   // reused as X2
    float* AEVb= ws + 6 * NB;

    const int MT = NROWS / 16;  // 800 row tiles

    embed_kernel<<<BT_TOTAL, 256, 0, stream>>>(
        states, enemy_states, actions, agent_emb, action_emb, enemy_emb,
        W_ally, b_ally, W_enemy, b_enemy, ln0_g, ln0_b, SE, SAE, AE);

    gemm_wmma<<<MT, 512, 0, stream>>>(SE,  W_q, b_q,     Qb,   0);
    gemm_wmma<<<MT, 512, 0, stream>>>(SE,  W_k, b_k,     Kb,   0);
    gemm_wmma<<<MT, 512, 0, stream>>>(SAE, W_v, b_v,     Vfb,  0);
    gemm_wmma<<<MT, 512, 0, stream>>>(AE,  W_v, nullptr, AEVb, 0);

    float* X1 = AE;  // AE dead from here
    attn_kernel<<<BT_TOTAL, 256, 0, stream>>>(
        Qb, Kb, Vfb, AEVb, SE, SAE, agent_masks, gumbel,
        W_h, b_h, ln1_g, ln1_b, X1);

    float* F1 = Qb;
    float* F2 = Kb;
    float* X2 = Vfb;
    float* XC = SE;
    gemm_wmma<<<MT, 512, 0, stream>>>(X1, W_f1, b_f1, F1, 1);   // gelu
    gemm_wmma<<<MT, 512, 0, stream>>>(F1, W_f2, b_f2, F2, 0);
    resln_kernel<<<NROWS, 256, 0, stream>>>(X1, F2, ln2_g, ln2_b, X2);
    gemm_wmma<<<MT, 512, 0, stream>>>(X2, W_c, b_c, XC, 1);     // gelu
    head_kernel<<<NROWS, 256, 0, stream>>>(XC, ln3_g, ln3_b, W_out, b_out, (float*)d_out);
}